// SparseAttention_82875688944377
// MI455X (gfx1250) — compile-verified
//
#include <hip/hip_runtime.h>
#include <math.h>

typedef float v2f __attribute__((ext_vector_type(2)));
typedef float v8f __attribute__((ext_vector_type(8)));

#define NROW 1024     // 4 * 256 query rows
#define NMEM 20000    // memory rows
#define DIM  128      // feature dim
#define KTOP 2000     // 10% of 20000

// ---------------------------------------------------------------------------
// Kernel 1: scores[r][n] = dot(query[r][:], memory[n][:])   (f32 WMMA GEMM)
// wave tile: 16 rows x 80 cols (5 v8f accumulators); block = 8 waves stacked
// over rows -> 128 rows x 80 cols per block; grid (8, 250) covers 1024x20000.
// ---------------------------------------------------------------------------
__global__ __launch_bounds__(256) void scores_wmma_kernel(
    const float* __restrict__ Q,   // [1024][128]
    const float* __restrict__ M,   // [20000][128]
    float* __restrict__ S)         // [1024][20000]
{
    const int wave = threadIdx.x >> 5;
    const int lane = threadIdx.x & 31;
    const int l16  = lane & 15;
    const int hi   = lane >> 4;            // 0: lanes 0-15, 1: lanes 16-31

    const int rowBase = blockIdx.x * 128 + wave * 16;
    const int colBase = blockIdx.y * 80;

    v8f c0 = {}, c1 = {}, c2 = {}, c3 = {}, c4 = {};

    // A (16x4 f32): lanes 0-15 hold K = k,k+1; lanes 16-31 hold K = k+2,k+3
    const float* qptr = Q + (size_t)(rowBase + l16) * DIM + 2 * hi;
    // B (4x16 f32): VGPR0 = K (k + 2*hi), VGPR1 = K (k + 2*hi + 1), N = l16
    const float* m0 = M + (size_t)(colBase +  0 + l16) * DIM + 2 * hi;
    const float* m1 = M + (size_t)(colBase + 16 + l16) * DIM + 2 * hi;
    const float* m2 = M + (size_t)(colBase + 32 + l16) * DIM + 2 * hi;
    const float* m3 = M + (size_t)(colBase + 48 + l16) * DIM + 2 * hi;
    const float* m4 = M + (size_t)(colBase + 64 + l16) * DIM + 2 * hi;

    #pragma unroll 4
    for (int k = 0; k < DIM; k += 4) {
        v2f a  = *(const v2f*)(qptr + k);
        v2f b0 = *(const v2f*)(m0 + k);
        v2f b1 = *(const v2f*)(m1 + k);
        v2f b2 = *(const v2f*)(m2 + k);
        v2f b3 = *(const v2f*)(m3 + k);
        v2f b4 = *(const v2f*)(m4 + k);
        c0 = __builtin_amdgcn_wmma_f32_16x16x4_f32(false, a, false, b0, (short)0, c0, false, false);
        c1 = __builtin_amdgcn_wmma_f32_16x16x4_f32(false, a, false, b1, (short)0, c1, false, false);
        c2 = __builtin_amdgcn_wmma_f32_16x16x4_f32(false, a, false, b2, (short)0, c2, false, false);
        c3 = __builtin_amdgcn_wmma_f32_16x16x4_f32(false, a, false, b3, (short)0, c3, false, false);
        c4 = __builtin_amdgcn_wmma_f32_16x16x4_f32(false, a, false, b4, (short)0, c4, false, false);
    }

    // C/D layout: VGPR v, lanes 0-15 -> M=v, lanes 16-31 -> M=v+8, N = l16
    const int srow = rowBase + 8 * hi;
    #pragma unroll
    for (int v = 0; v < 8; ++v) {
        size_t base = (size_t)(srow + v) * NMEM + colBase + l16;
        S[base +  0] = c0[v];
        S[base + 16] = c1[v];
        S[base + 32] = c2[v];
        S[base + 48] = c3[v];
        S[base + 64] = c4[v];
    }
}

// ---------------------------------------------------------------------------
// Order-preserving f32 <-> u32 key transform (larger key <=> larger float)
// ---------------------------------------------------------------------------
__device__ __forceinline__ unsigned f2key(float f) {
    unsigned u = __float_as_uint(f);
    return (u & 0x80000000u) ? ~u : (u | 0x80000000u);
}
__device__ __forceinline__ float key2f(unsigned u) {
    return __uint_as_float((u & 0x80000000u) ? (u & 0x7FFFFFFFu) : ~u);
}

// ---------------------------------------------------------------------------
// Kernel 2: exact k-th-largest (k = 2000) per row via MSB-first radix select,
// one block (256 threads) per row. Also produces the row max.
// ---------------------------------------------------------------------------
__global__ __launch_bounds__(256) void radix_select_kernel(
    const float* __restrict__ S,     // [1024][20000]
    float* __restrict__ thr,         // [1024]
    float* __restrict__ rowmax)      // [1024]
{
    const int row = blockIdx.x;
    const int tid = threadIdx.x;
    const float* srow = S + (size_t)row * NMEM;

    __shared__ unsigned hist[256];
    __shared__ unsigned sh_sel, sh_krem, sh_max;

    if (tid == 0) { sh_krem = KTOP; sh_max = 0u; }

    unsigned prefix = 0u, mask = 0u;

    for (int shift = 24; shift >= 0; shift -= 8) {
        __syncthreads();
        hist[tid] = 0u;
        __syncthreads();

        for (int n = tid; n < NMEM; n += 256) {
            unsigned u = f2key(srow[n]);
            if (shift == 24) {
                atomicMax(&sh_max, u);
                atomicAdd(&hist[u >> 24], 1u);
            } else if ((u & mask) == prefix) {
                atomicAdd(&hist[(u >> shift) & 255u], 1u);
            }
        }
        __syncthreads();

        if (tid == 0) {
            unsigned krem = sh_krem, cum = 0u;
            int b = 255;
            for (; b >= 0; --b) {
                cum += hist[b];
                if (cum >= krem) break;
            }
            if (b < 0) b = 0;                       // can't happen; safety
            sh_sel  = (unsigned)b;
            sh_krem = krem - (cum - hist[b]);
        }
        __syncthreads();

        prefix |= (sh_sel << shift);
        mask   |= (0xFFu << shift);
    }

    if (tid == 0) {
        thr[row]    = key2f(prefix);   // exact 2000th-largest score
        rowmax[row] = key2f(sh_max);   // row max (== max of the top-k)
    }
}

// ---------------------------------------------------------------------------
// Kernel 3: out[row][d] = sum_{s>=t} e^(s-max) * memory[n][d] / sum e^(s-max)
// One block of 128 threads per row; thread = feature dim d. Weights for a
// 128-wide tile of n are staged in LDS; ~90% of memory-row reads are skipped.
// ---------------------------------------------------------------------------
__global__ __launch_bounds__(128) void weighted_sum_kernel(
    const float* __restrict__ S,       // [1024][20000]
    const float* __restrict__ M,       // [20000][128]
    const float* __restrict__ thr,     // [1024]
    const float* __restrict__ rowmax,  // [1024]
    float* __restrict__ out)           // [1024][128]
{
    const int row = blockIdx.x;
    const int tid = threadIdx.x;       // d index
    const float t = thr[row];
    const float m = rowmax[row];
    const float* srow = S + (size_t)row * NMEM;

    __shared__ float wts[128];
    __shared__ float red[128];

    float acc = 0.0f;
    float wsum = 0.0f;

    for (int base = 0; base < NMEM; base += 128) {
        int n = base + tid;
        float w = 0.0f;
        if (n < NMEM) {
            float s = srow[n];
            if (s >= t) w = __expf(s - m);
        }
        wts[tid] = w;
        wsum += w;
        __syncthreads();

        #pragma unroll 4
        for (int j = 0; j < 128; ++j) {
            float wj = wts[j];
            if (wj > 0.0f)
                acc += wj * M[(size_t)(base + j) * DIM + tid];
        }
        __syncthreads();
    }

    red[tid] = wsum;
    __syncthreads();
    #pragma unroll
    for (int off = 64; off > 0; off >>= 1) {
        if (tid < off) red[tid] += red[tid + off];
        __syncthreads();
    }

    out[(size_t)row * DIM + tid] = acc / red[0];
}

// ---------------------------------------------------------------------------
extern "C" void kernel_launch(void* const* d_in, const int* in_sizes, int n_in,
                              void* d_out, int out_size, void* d_ws, size_t ws_size,
                              hipStream_t stream) {
    const float* Q = (const float*)d_in[0];   // query  [4,256,128] -> [1024][128]
    const float* M = (const float*)d_in[1];   // memory [20000][128]
    float* out = (float*)d_out;               // [1024][128]

    float* scores = (float*)d_ws;                       // 1024*20000 f32 = 80 MB
    float* thr    = scores + (size_t)NROW * NMEM;       // 1024 f32
    float* rmax   = thr + NROW;                         // 1024 f32

    dim3 g1(NROW / 128, NMEM / 80);                     // (8, 250)
    scores_wmma_kernel<<<g1, dim3(256), 0, stream>>>(Q, M, scores);
    radix_select_kernel<<<dim3(NROW), dim3(256), 0, stream>>>(scores, thr, rmax);
    weighted_sum_kernel<<<dim3(NROW), dim3(128), 0, stream>>>(scores, M, thr, rmax, out);
}